// GAT_NET_51238959841810
// MI455X (gfx1250) — compile-verified
//
#include <hip/hip_runtime.h>
#include <hip/hip_bf16.h>

#define FEAT 256
#define HID 64
#define HEADS 4
#define NEG_SLOPE 0.2f

typedef __attribute__((ext_vector_type(2))) float v2f;
typedef __attribute__((ext_vector_type(8))) float v8f;

// ---- monotone float <-> unsigned key (supports negatives; 0 == below -inf) ----
__device__ __forceinline__ unsigned f2key(float f) {
    int i = __float_as_int(f);
    return (unsigned)(i ^ ((i >> 31) | 0x80000000));
}
__device__ __forceinline__ float key2f(unsigned k) {
    int i = (k & 0x80000000u) ? (int)(k ^ 0x80000000u) : ~((int)k);
    return __int_as_float(i);
}

// edge e in [0, E+N): first E from edge_index, last N are self loops
__device__ __forceinline__ void edge_nodes(const int* __restrict__ ei, int E, int e, int& s, int& d) {
    if (e < E) { s = ei[e]; d = ei[E + e]; }
    else       { int n = e - E; s = n; d = n; }
}

// -------------------- init accumulators (re-run every call: deterministic) ------
__global__ void k_init(float* __restrict__ out1, unsigned* __restrict__ emax1,
                       float* __restrict__ den1, unsigned* __restrict__ emax2,
                       float* __restrict__ den2, float* __restrict__ out,
                       const float* __restrict__ b2, int N) {
    long long i = (long long)blockIdx.x * 256 + threadIdx.x;
    long long tot = (long long)N * FEAT;
    if (i < tot) out1[i] = 0.0f;
    if (i < (long long)N * HEADS) { emax1[i] = 0u; den1[i] = 0.0f; }
    if (i < N) { emax2[i] = 0u; den2[i] = 0.0f; out[i] = b2[0]; }
}

// ---- GEMM1: xw = x @ w1, f32 WMMA 16x16x4; wave computes 16x64 (4 col tiles) ---
// Row index is clamped (not branched) so the K-loop is exec-mask-stable; the
// 4 accumulators are independent, so the 4 v_wmma per K-step co-execute and
// overlap with the next step's B loads instead of serializing on one C/D.
__global__ void __launch_bounds__(32)
k_gemm1(const float* __restrict__ X, const float* __restrict__ W,
        float* __restrict__ XW, int N) {
    const int lane    = threadIdx.x;
    const int m16     = lane & 15;
    const int khalf   = lane >> 4;                  // 0: K={0,1}, 1: K={2,3}
    const int rowBase = blockIdx.x * 16;
    const int row     = min(rowBase + m16, N - 1);  // clamp: dup rows, guarded store
    const int col     = blockIdx.y * 64 + m16;      // first of 4 col tiles

    v8f acc0 = {}, acc1 = {}, acc2 = {}, acc3 = {};
    const float* xrow  = X + (size_t)row * FEAT;
    const float* wbase = W + col;
    for (int k = 0; k < FEAT; k += 4) {
        v2f a = *(const v2f*)(xrow + k + khalf * 2);            // 8B-aligned b64 load
        const float* w0 = wbase + (size_t)(k + khalf * 2) * (HEADS * HID);
        const float* w1 = w0 + (HEADS * HID);
        v2f b0, b1, b2, b3;
        b0.x = w0[0];  b0.y = w1[0];
        b1.x = w0[16]; b1.y = w1[16];
        b2.x = w0[32]; b2.y = w1[32];
        b3.x = w0[48]; b3.y = w1[48];
        acc0 = __builtin_amdgcn_wmma_f32_16x16x4_f32(false, a, false, b0, (short)0, acc0, false, false);
        acc1 = __builtin_amdgcn_wmma_f32_16x16x4_f32(false, a, false, b1, (short)0, acc1, false, false);
        acc2 = __builtin_amdgcn_wmma_f32_16x16x4_f32(false, a, false, b2, (short)0, acc2, false, false);
        acc3 = __builtin_amdgcn_wmma_f32_16x16x4_f32(false, a, false, b3, (short)0, acc3, false, false);
    }
#pragma unroll
    for (int r = 0; r < 8; ++r) {
        int orow = rowBase + r + khalf * 8;         // D: M = r + 8*(lane/16)
        if (orow < N) {
            float* op = XW + (size_t)orow * (HEADS * HID) + col;
            op[0]  = acc0[r];
            op[16] = acc1[r];
            op[32] = acc2[r];
            op[48] = acc3[r];
        }
    }
}

// -------------------- per-node attention logits: alpha_s/alpha_d [N,H] ---------
__global__ void k_alpha1(const float* __restrict__ XW, const float* __restrict__ asrc,
                         const float* __restrict__ adst, float* __restrict__ alS,
                         float* __restrict__ alD, int N) {
    int i = blockIdx.x * blockDim.x + threadIdx.x;   // flat (n,h)
    if (i >= N * HEADS) return;
    int h = i & (HEADS - 1);
    const float* v  = XW + (size_t)(i >> 2) * FEAT + h * HID;
    const float* as = asrc + h * HID;
    const float* ad = adst + h * HID;
    float ss = 0.0f, sd = 0.0f;
    for (int j = 0; j < HID; j += 4) {
        float4 xv = *(const float4*)(v + j);
        float4 a1 = *(const float4*)(as + j);
        float4 a2 = *(const float4*)(ad + j);
        ss += xv.x * a1.x + xv.y * a1.y + xv.z * a1.z + xv.w * a1.w;
        sd += xv.x * a2.x + xv.y * a2.y + xv.z * a2.z + xv.w * a2.w;
    }
    alS[i] = ss; alD[i] = sd;
}

// -------------------- layer1 edge pass 1: e = leaky(as[src]+ad[dst]); seg-max ---
__global__ void k_edge_max1(const int* __restrict__ ei, int E, int N,
                            const float* __restrict__ alS, const float* __restrict__ alD,
                            float* __restrict__ ebuf, unsigned* __restrict__ emax) {
    int e = blockIdx.x * blockDim.x + threadIdx.x;
    if (e >= E + N) return;
    int s, d; edge_nodes(ei, E, e, s, d);
    float4 a = *(const float4*)(alS + (size_t)s * HEADS);
    float4 b = *(const float4*)(alD + (size_t)d * HEADS);
    float4 r;
    r.x = a.x + b.x; r.y = a.y + b.y; r.z = a.z + b.z; r.w = a.w + b.w;
    r.x = r.x > 0.0f ? r.x : NEG_SLOPE * r.x;
    r.y = r.y > 0.0f ? r.y : NEG_SLOPE * r.y;
    r.z = r.z > 0.0f ? r.z : NEG_SLOPE * r.z;
    r.w = r.w > 0.0f ? r.w : NEG_SLOPE * r.w;
    *(float4*)(ebuf + (size_t)e * HEADS) = r;
    unsigned* m = emax + (size_t)d * HEADS;
    atomicMax(m + 0, f2key(r.x));
    atomicMax(m + 1, f2key(r.y));
    atomicMax(m + 2, f2key(r.z));
    atomicMax(m + 3, f2key(r.w));
}

// -------------------- convert max keys -> floats in place ----------------------
__global__ void k_keyconv(unsigned* __restrict__ buf, long long n) {
    long long i = (long long)blockIdx.x * 256 + threadIdx.x;
    if (i >= n) return;
    unsigned k = buf[i];
    float f = (k == 0u) ? 0.0f : key2f(k);   // untouched => no edges => 0 (ref semantics)
    ((float*)buf)[i] = f;
}

// -------------------- layer1 edge pass 2: p = exp(e - max[dst]); seg-sum -------
__global__ void k_edge_exp1(const int* __restrict__ ei, int E, int N,
                            float* __restrict__ ebuf, const float* __restrict__ emaxf,
                            float* __restrict__ den) {
    int e = blockIdx.x * blockDim.x + threadIdx.x;
    if (e >= E + N) return;
    int s, d; edge_nodes(ei, E, e, s, d);
    float4 ev = *(const float4*)(ebuf + (size_t)e * HEADS);
    float4 mx = *(const float4*)(emaxf + (size_t)d * HEADS);
    float4 p;
    p.x = expf(ev.x - mx.x); p.y = expf(ev.y - mx.y);
    p.z = expf(ev.z - mx.z); p.w = expf(ev.w - mx.w);
    *(float4*)(ebuf + (size_t)e * HEADS) = p;
    float* dn = den + (size_t)d * HEADS;
    unsafeAtomicAdd(dn + 0, p.x);
    unsafeAtomicAdd(dn + 1, p.y);
    unsafeAtomicAdd(dn + 2, p.z);
    unsafeAtomicAdd(dn + 3, p.w);
}

// -------------------- layer1 aggregate: wave per edge, lane owns 8 feats -------
__global__ void k_edge_agg1(const int* __restrict__ ei, int E, int N,
                            const float* __restrict__ pbuf, const float* __restrict__ den,
                            const float* __restrict__ XW, float* __restrict__ out1) {
    int e = blockIdx.x * 8 + (threadIdx.x >> 5);
    if (e >= E + N) return;
    int lane = threadIdx.x & 31;
    int s, d; edge_nodes(ei, E, e, s, d);
    int h = lane >> 3;                                    // 8 lanes per head (8*8=64)
    float alpha = pbuf[(size_t)e * HEADS + h] /
                  (den[(size_t)d * HEADS + h] + 1e-16f);
    const float* src = XW  + (size_t)s * FEAT + lane * 8;
    float*       dst = out1 + (size_t)d * FEAT + lane * 8;
    float4 v0 = *(const float4*)(src);
    float4 v1 = *(const float4*)(src + 4);
    unsafeAtomicAdd(dst + 0, v0.x * alpha);
    unsafeAtomicAdd(dst + 1, v0.y * alpha);
    unsafeAtomicAdd(dst + 2, v0.z * alpha);
    unsafeAtomicAdd(dst + 3, v0.w * alpha);
    unsafeAtomicAdd(dst + 4, v1.x * alpha);
    unsafeAtomicAdd(dst + 5, v1.y * alpha);
    unsafeAtomicAdd(dst + 6, v1.z * alpha);
    unsafeAtomicAdd(dst + 7, v1.w * alpha);
}

// ------------ layer2 projection: z[n] = sum relu(out1[n]+b1)*w2 (wave/node) ----
__global__ void k_proj2(const float* __restrict__ out1, const float* __restrict__ b1,
                        const float* __restrict__ w2, float* __restrict__ z, int N) {
    int n = blockIdx.x * 8 + (threadIdx.x >> 5);
    if (n >= N) return;
    int lane = threadIdx.x & 31;
    const float* row = out1 + (size_t)n * FEAT + lane * 8;
    const float* bb  = b1 + lane * 8;
    const float* ww  = w2 + lane * 8;
    float sum = 0.0f;
#pragma unroll
    for (int j = 0; j < 8; ++j) {
        float v = row[j] + bb[j];
        v = v > 0.0f ? v : 0.0f;
        sum += v * ww[j];
    }
    for (int off = 16; off; off >>= 1) sum += __shfl_xor(sum, off, 32);
    if (lane == 0) z[n] = sum;
}

// -------------------- layer2 edge passes (H=1, F=1, scalar) --------------------
__global__ void k_edge_max2(const int* __restrict__ ei, int E, int N,
                            const float* __restrict__ z, const float* __restrict__ as2,
                            const float* __restrict__ ad2, float* __restrict__ ebuf,
                            unsigned* __restrict__ emax) {
    int e = blockIdx.x * blockDim.x + threadIdx.x;
    if (e >= E + N) return;
    int s, d; edge_nodes(ei, E, e, s, d);
    float v = z[s] * as2[0] + z[d] * ad2[0];
    v = v > 0.0f ? v : NEG_SLOPE * v;
    ebuf[e] = v;
    atomicMax(&emax[d], f2key(v));
}

__global__ void k_edge_exp2(const int* __restrict__ ei, int E, int N,
                            float* __restrict__ ebuf, const float* __restrict__ emaxf,
                            float* __restrict__ den) {
    int e = blockIdx.x * blockDim.x + threadIdx.x;
    if (e >= E + N) return;
    int s, d; edge_nodes(ei, E, e, s, d);
    float p = expf(ebuf[e] - emaxf[d]);
    ebuf[e] = p;
    unsafeAtomicAdd(&den[d], p);
}

__global__ void k_edge_agg2(const int* __restrict__ ei, int E, int N,
                            const float* __restrict__ ebuf, const float* __restrict__ den,
                            const float* __restrict__ z, float* __restrict__ out) {
    int e = blockIdx.x * blockDim.x + threadIdx.x;
    if (e >= E + N) return;
    int s, d; edge_nodes(ei, E, e, s, d);
    unsafeAtomicAdd(&out[d], z[s] * ebuf[e] / (den[d] + 1e-16f));
}

// ==============================================================================
extern "C" void kernel_launch(void* const* d_in, const int* in_sizes, int n_in,
                              void* d_out, int out_size, void* d_ws, size_t ws_size,
                              hipStream_t stream) {
    const float* x     = (const float*)d_in[0];
    const int*   ei    = (const int*)d_in[1];
    const float* w1    = (const float*)d_in[2];
    const float* as1   = (const float*)d_in[3];
    const float* ad1   = (const float*)d_in[4];
    const float* b1    = (const float*)d_in[5];
    const float* w2    = (const float*)d_in[6];
    const float* as2   = (const float*)d_in[7];
    const float* ad2   = (const float*)d_in[8];
    const float* b2    = (const float*)d_in[9];
    float*       out   = (float*)d_out;

    const int N  = in_sizes[0] / FEAT;
    const int E  = in_sizes[1] / 2;
    const int E2 = E + N;

    // workspace carving (floats); total ~113 MB -> fits L2-resident working set
    float* ws = (float*)d_ws;
    size_t o = 0;
    float*    xw    = ws + o; o += (size_t)N * FEAT;
    float*    out1  = ws + o; o += (size_t)N * FEAT;
    float*    alS   = ws + o; o += (size_t)N * HEADS;
    float*    alD   = ws + o; o += (size_t)N * HEADS;
    unsigned* emax1 = (unsigned*)(ws + o); o += (size_t)N * HEADS;
    float*    den1  = ws + o; o += (size_t)N * HEADS;
    float*    ebuf1 = ws + o; o += (size_t)E2;
    float*    z     = ws + o; o += (size_t)N;
    unsigned* emax2 = (unsigned*)(ws + o); o += (size_t)N;
    float*    den2  = ws + o; o += (size_t)N;
    float*    ebuf2 = ws + o; o += (size_t)E2;

    const dim3 B(256);
    long long totInit = (long long)N * FEAT;

    k_init<<<(unsigned)((totInit + 255) / 256), B, 0, stream>>>(
        out1, emax1, den1, emax2, den2, out, b2, N);

    // 16 rows x 64 cols per wave: grid ((N+15)/16, 256/64)
    k_gemm1<<<dim3((N + 15) / 16, (HEADS * HID) / 64), dim3(32), 0, stream>>>(x, w1, xw, N);

    k_alpha1<<<(N * HEADS + 255) / 256, B, 0, stream>>>(xw, as1, ad1, alS, alD, N);

    k_edge_max1<<<(E2 + 255) / 256, B, 0, stream>>>(ei, E, N, alS, alD, ebuf1, emax1);
    k_keyconv<<<(N * HEADS + 255) / 256, B, 0, stream>>>(emax1, (long long)N * HEADS);
    k_edge_exp1<<<(E2 + 255) / 256, B, 0, stream>>>(ei, E, N, ebuf1, (const float*)emax1, den1);
    k_edge_agg1<<<(E2 + 7) / 8, B, 0, stream>>>(ei, E, N, ebuf1, den1, xw, out1);

    k_proj2<<<(N + 7) / 8, B, 0, stream>>>(out1, b1, w2, z, N);

    k_edge_max2<<<(E2 + 255) / 256, B, 0, stream>>>(ei, E, N, z, as2, ad2, ebuf2, emax2);
    k_keyconv<<<(N + 255) / 256, B, 0, stream>>>(emax2, (long long)N);
    k_edge_exp2<<<(E2 + 255) / 256, B, 0, stream>>>(ei, E, N, ebuf2, (const float*)emax2, den2);
    k_edge_agg2<<<(E2 + 255) / 256, B, 0, stream>>>(ei, E, N, ebuf2, den2, z, out);
}